// Loss_41910290874694
// MI455X (gfx1250) — compile-verified
//
#include <hip/hip_runtime.h>
#include <math.h>

// ---------------- problem constants ----------------
#define B_      8
#define NP_     32768
#define KPRIM_  128
#define NSKEL_  2048          // = KPRIM_ * 16 curve samples
#define NTOT_   (B_ * NP_)    // 262144
#define NROWS_  (B_ * NP_)    // overlap rows of 128 floats

typedef float v2f __attribute__((ext_vector_type(2)));
typedef float v8f __attribute__((ext_vector_type(8)));

// ---------------- init workspace (runs every call; graph-replay safe) ----
__global__ void init_kernel(float* acc, float* minA, float* minB) {
    int i = blockIdx.x * blockDim.x + threadIdx.x;
    if (i < 16) acc[i] = 0.0f;
    int stride = gridDim.x * blockDim.x;
    for (int j = i; j < B_ * NSKEL_; j += stride) {
        minA[j] = INFINITY;
        minB[j] = INFINITY;
    }
}

// ---------------- recon: sum (p-g)^2 ----------------
__global__ void recon_kernel(const float* __restrict__ p,
                             const float* __restrict__ g,
                             float* __restrict__ acc) {
    __shared__ float sacc[8];
    const float4* p4 = (const float4*)p;
    const float4* g4 = (const float4*)g;
    int idx    = blockIdx.x * blockDim.x + threadIdx.x;
    int stride = gridDim.x * blockDim.x;
    float local = 0.0f;
    for (int i = idx; i < NTOT_ / 4; i += stride) {
        float4 a = p4[i], b = g4[i];
        float dx = a.x - b.x, dy = a.y - b.y, dz = a.z - b.z, dw = a.w - b.w;
        local += dx * dx + dy * dy + dz * dz + dw * dw;
    }
    for (int off = 16; off; off >>= 1) local += __shfl_xor(local, off, 32);
    int lane = threadIdx.x & 31, wave = threadIdx.x >> 5;
    if (lane == 0) sacc[wave] = local;
    __syncthreads();
    if (threadIdx.x == 0) {
        float t = 0.0f;
        for (int i = 0; i < 8; ++i) t += sacc[i];
        atomicAdd(acc, t);
    }
}

// ---------------- overlap: one wave per 128-float row (128 MB stream) ----
#define ROWS_PER_WAVE 32
__global__ void overlap_kernel(const float* __restrict__ occ,
                               float* __restrict__ acc) {
    __shared__ float sacc[8];
    int lane = threadIdx.x & 31, wave = threadIdx.x >> 5;
    int gw   = blockIdx.x * 8 + wave;           // global wave id
    const float4* occ4 = (const float4*)occ;
    float local = 0.0f;
    int row0 = gw * ROWS_PER_WAVE;
    for (int r = 0; r < ROWS_PER_WAVE; ++r) {
        int row = row0 + r;                      // row of 128 floats (32 float4)
        // prefetch 8 rows (4 KB) ahead, clamped in-bounds
        int prow = row + 8;
        if (prow >= NROWS_) prow = NROWS_ - 1;
        __builtin_prefetch(&occ4[prow * 32 + lane], 0, 3);
        float4 v = occ4[row * 32 + lane];        // fully coalesced 512B/wave
        float s = (v.x > 0.5f ? v.x : 0.0f) + (v.y > 0.5f ? v.y : 0.0f) +
                  (v.z > 0.5f ? v.z : 0.0f) + (v.w > 0.5f ? v.w : 0.0f);
        for (int off = 16; off; off >>= 1) s += __shfl_xor(s, off, 32);
        local += fmaxf(s - 32.0f, 0.0f);         // int(K/4) = 32
    }
    if (lane == 0) sacc[wave] = local;
    __syncthreads();
    if (threadIdx.x == 0) {
        float t = 0.0f;
        for (int i = 0; i < 8; ++i) t += sacc[i];
        atomicAdd(acc, t);
    }
}

// ---------------- curves + sigmoid sum: 1 thread per (b,k) ----------------
__global__ void curves_kernel(const float* __restrict__ params,
                              const float* __restrict__ wts,
                              const float* __restrict__ basis,
                              float* __restrict__ curves,
                              float* __restrict__ accSig) {
    int t = blockIdx.x * blockDim.x + threadIdx.x;   // 0..1023
    int b = t >> 7, k = t & 127;
    float w   = wts[b * KPRIM_ + k];
    float sig = 1.0f / (1.0f + expf(-w));
    float s = sig;
    for (int off = 16; off; off >>= 1) s += __shfl_xor(s, off, 32);
    if ((threadIdx.x & 31) == 0) atomicAdd(accSig, s);

    float cp[12];
    const float* pp = params + (b * KPRIM_ + k) * 16;
#pragma unroll
    for (int j = 0; j < 12; ++j) cp[j] = pp[j] * w;
    float* cout = curves + (b * NSKEL_ + k * 16) * 3;
#pragma unroll
    for (int tt = 0; tt < 16; ++tt) {
        float b0 = basis[tt * 4 + 0], b1 = basis[tt * 4 + 1];
        float b2 = basis[tt * 4 + 2], b3 = basis[tt * 4 + 3];
#pragma unroll
        for (int d = 0; d < 3; ++d)
            cout[tt * 3 + d] = b0 * cp[d] + b1 * cp[3 + d] +
                               b2 * cp[6 + d] + b3 * cp[9 + d];
    }
}

// ---------------- chamfer min via V_WMMA_F32_16X16X4_F32 ----------------
// One wave owns a 16-point tile of P, scans all 2048 points of Q.
// f32 16x4 A layout (ISA 7.12.2): lane l holds A[M=l%16][K=2*(l/16)+v] in vgpr v.
// We exploit the spare K=3 slot to compute |q|^2 - 2*p.q in a single matrix op:
//   A rows:  (-2x, -2y, -2z, 1)   B cols: (qx, qy, qz, |q|^2)
//   D[m][n] = |q_n|^2 - 2 * p_m . q_n
// The constant |p_m|^2 add and the max(.,0) clamp commute with min -> epilogue.
__global__ void __launch_bounds__(256)
chamfer_kernel(const float* __restrict__ P, const float* __restrict__ Q,
               float* __restrict__ outMin) {
    int lane = threadIdx.x & 31;
    int gw   = blockIdx.x * (blockDim.x >> 5) + (threadIdx.x >> 5); // 0..1023
    int b    = gw >> 7;
    int tile = gw & 127;
    int hi   = lane >> 4;
    int lo   = lane & 15;

    const float* pp = P + (b * NSKEL_ + tile * 16 + lo) * 3;
    float px = pp[0], py = pp[1], pz = pp[2];
    v2f a;
    a.x = hi ? -2.0f * pz : -2.0f * px;
    a.y = hi ? 1.0f : -2.0f * py;

    // |p|^2 per lane (each lane loaded all 3 coords), broadcast per row
    float a2 = px * px + py * py + pz * pz;       // lane l holds |p[l%16]|^2
    float a2row[8];
#pragma unroll
    for (int r = 0; r < 8; ++r) a2row[r] = __shfl(a2, r + 8 * hi, 32);

    float mn[8];
#pragma unroll
    for (int r = 0; r < 8; ++r) mn[r] = INFINITY;

    for (int mt = 0; mt < NSKEL_ / 16; ++mt) {
        const float* qq = Q + (b * NSKEL_ + mt * 16 + lo) * 3;
        float qx = qq[0], qy = qq[1], qz = qq[2];
        float q2 = qx * qx + qy * qy + qz * qz;
        v2f bv;
        bv.x = hi ? qz : qx;
        bv.y = hi ? q2 : qy;

        v8f c = {};
        // D[m][n] = |q_n|^2 - 2 * p_m . q_n   (one matrix op, K=4)
        v8f d = __builtin_amdgcn_wmma_f32_16x16x4_f32(
            false, a, false, bv, (short)0, c, false, false);
#pragma unroll
        for (int r = 0; r < 8; ++r) mn[r] = fminf(mn[r], d[r]);
    }

    // min across the 16 lanes of each half (xor 1,2,4,8 stays within a half)
#pragma unroll
    for (int r = 0; r < 8; ++r) {
        float v = mn[r];
        for (int off = 8; off; off >>= 1) v = fminf(v, __shfl_xor(v, off, 32));
        mn[r] = fmaxf(a2row[r] + v, 0.0f);        // clamped min d^2
    }
    if (lo == 0) {                                 // lane0 -> rows 0..7, lane16 -> rows 8..15
        float* o = outMin + b * NSKEL_ + tile * 16 + hi * 8;
#pragma unroll
        for (int r = 0; r < 8; ++r) o[r] = mn[r];
    }
}

// ---------------- final assembly ----------------
__global__ void finalize_kernel(const float* __restrict__ acc,
                                const float* __restrict__ minA,
                                const float* __restrict__ minB,
                                float* __restrict__ out) {
    __shared__ float sdata[256];
    int tid = threadIdx.x;
    float scalar = acc[0] * (1.0f / (float)NTOT_)            // recon (scale 1.0)
                 + 0.1f * acc[1] * (1.0f / (float)NTOT_)     // overlap mean * 0.1
                 + 0.1f * sqrtf(acc[2]);                     // parsimony
    for (int b = 0; b < B_; ++b) {
        float local = 0.0f;
        for (int i = tid; i < NSKEL_; i += 256)
            local += sqrtf(minA[b * NSKEL_ + i]) + sqrtf(minB[b * NSKEL_ + i]);
        sdata[tid] = local;
        __syncthreads();
        for (int s = 128; s > 0; s >>= 1) {
            if (tid < s) sdata[tid] += sdata[tid + s];
            __syncthreads();
        }
        if (tid == 0) out[b] = scalar + sdata[0] * (1.0f / (float)NSKEL_);
        __syncthreads();
    }
}

extern "C" void kernel_launch(void* const* d_in, const int* in_sizes, int n_in,
                              void* d_out, int out_size, void* d_ws, size_t ws_size,
                              hipStream_t stream) {
    const float* predict = (const float*)d_in[0];   // (8, 32768)
    const float* gt      = (const float*)d_in[1];   // (8, 32768)
    const float* occ     = (const float*)d_in[2];   // (8, 32768, 128)
    const float* params  = (const float*)d_in[3];   // (8, 128, 16)
    const float* wts     = (const float*)d_in[4];   // (8, 128)
    const float* skel    = (const float*)d_in[5];   // (8, 2048, 3)
    const float* basis   = (const float*)d_in[6];   // (16, 4)
    float* out = (float*)d_out;                     // (8,)

    float* acc    = (float*)d_ws;        // [0]=recon sum, [1]=overlap sum, [2]=sigmoid sum
    float* curves = acc + 16;            // 8*2048*3 = 49152 floats
    float* minA   = curves + B_ * NSKEL_ * 3;   // 16384 floats (min d^2, skel->curves)
    float* minB   = minA + B_ * NSKEL_;         // 16384 floats (min d^2, curves->skel)

    init_kernel   <<<64,   256, 0, stream>>>(acc, minA, minB);
    recon_kernel  <<<64,   256, 0, stream>>>(predict, gt, acc + 0);
    overlap_kernel<<<NROWS_ / (8 * ROWS_PER_WAVE), 256, 0, stream>>>(occ, acc + 1);
    curves_kernel <<<4,    256, 0, stream>>>(params, wts, basis, curves, acc + 2);
    chamfer_kernel<<<128,  256, 0, stream>>>(skel, curves, minA);
    chamfer_kernel<<<128,  256, 0, stream>>>(curves, skel, minB);
    finalize_kernel<<<1,   256, 0, stream>>>(acc, minA, minB, out);
}